// KHopSGC_54485955117400
// MI455X (gfx1250) — compile-verified
//
#include <hip/hip_runtime.h>
#include <hip/hip_bf16.h>

typedef __attribute__((ext_vector_type(2))) float v2f;
typedef __attribute__((ext_vector_type(8))) float v8f;

#define FDIM 128
#define ODIM 128
#define KHOP 3
#define KF   (KHOP * FDIM)

// ---------------------------------------------------------------------------
// Kernel 1: out[n, o] = b[o]  (float4 vectorized)
// ---------------------------------------------------------------------------
__global__ __launch_bounds__(256) void khop_init_out(float* __restrict__ out,
                                                     const float* __restrict__ b,
                                                     int total4) {
    int t = blockIdx.x * blockDim.x + threadIdx.x;   // one float4 per thread
    if (t >= total4) return;
    int q = t & (ODIM / 4 - 1);                      // which float4 within the row
    float4 bv = ((const float4*)b)[q];
    ((float4*)out)[t] = bv;
}

// ---------------------------------------------------------------------------
// Kernel 2: y = x @ W_k^T   ([N,128] x [128,128]), full fp32 via
// V_WMMA_F32_16X16X4_F32.  Block = 256 threads = 8 waves; blockIdx.x selects a
// 16-row strip of x; wave w computes the 16x16 tile for output columns
// [16w, 16w+16).  K loop: 128 in steps of 4 (32 WMMA ops per tile).
// EXEC is all-ones: N = 6250*16 exactly, no divergence (WMMA requirement).
// ---------------------------------------------------------------------------
__global__ __launch_bounds__(256) void khop_gemm_xWk(const float* __restrict__ x,
                                                     const float* __restrict__ W,
                                                     float* __restrict__ y,
                                                     int hop) {
    const int wave = threadIdx.x >> 5;       // 0..7 -> o-tile
    const int lane = threadIdx.x & 31;
    const int m0   = blockIdx.x * 16;
    const int o0   = wave * 16;

    const int half  = lane >> 4;             // 0: K even pair base 0, 1: base 2
    const int l15   = lane & 15;
    const int khalf = half * 2;

    // A: row (m0 + l15) of x;   B: row (o0 + l15) of W_k (= W[., hop*F + .])
    const float* xr = x + (size_t)(m0 + l15) * FDIM;
    const float* wr = W + (size_t)(o0 + l15) * KF + hop * FDIM;

    v8f c = {};
#pragma unroll 8
    for (int kk = 0; kk < FDIM; kk += 4) {
        v2f a = *(const v2f*)(xr + kk + khalf);   // {K=kk+2*half, K+1}
        v2f bm = *(const v2f*)(wr + kk + khalf);
        c = __builtin_amdgcn_wmma_f32_16x16x4_f32(
                /*neg_a=*/false, a, /*neg_b=*/false, bm,
                /*c_mod=*/(short)0, c, /*reuse_a=*/false, /*reuse_b=*/false);
    }

    // D layout: VGPR r <-> (m = r + 8*half, n = l15)
    const int mbase = m0 + half * 8;
    float* yo = y + (size_t)mbase * ODIM + o0 + l15;
#pragma unroll
    for (int r = 0; r < 8; ++r) {
        yo[(size_t)r * ODIM] = c[r];
    }
}

// ---------------------------------------------------------------------------
// Kernel 3: out[dst[e], :] += vals[e] * y[src[e], :]
// One wave per edge; each lane owns 4 contiguous floats (128-bit gather of the
// L2-resident y row, then 4 hardware fp32 atomics into the L2-resident out).
// Index/value loads hit one address per wave -> broadcast by the memory path.
// ---------------------------------------------------------------------------
__global__ __launch_bounds__(256) void khop_scatter(const int* __restrict__ src,
                                                    const int* __restrict__ dst,
                                                    const float* __restrict__ vals,
                                                    const float* __restrict__ y,
                                                    float* out, int E) {
    int t    = blockIdx.x * blockDim.x + threadIdx.x;
    int e    = t >> 5;
    int lane = t & 31;
    if (e >= E) return;

    int   s = src[e];
    int   d = dst[e];
    float v = vals[e];

    float4 m = ((const float4*)(y + (size_t)s * FDIM))[lane];
    float* op = out + (size_t)d * ODIM + lane * 4;

    __hip_atomic_fetch_add(op + 0, v * m.x, __ATOMIC_RELAXED, __HIP_MEMORY_SCOPE_AGENT);
    __hip_atomic_fetch_add(op + 1, v * m.y, __ATOMIC_RELAXED, __HIP_MEMORY_SCOPE_AGENT);
    __hip_atomic_fetch_add(op + 2, v * m.z, __ATOMIC_RELAXED, __HIP_MEMORY_SCOPE_AGENT);
    __hip_atomic_fetch_add(op + 3, v * m.w, __ATOMIC_RELAXED, __HIP_MEMORY_SCOPE_AGENT);
}

// ---------------------------------------------------------------------------
// Launch: init out with bias, then per hop: GEMM y_k into d_ws (51.2 MB),
// scatter edges of hop k.  Stream-ordered, so d_ws reuse across hops is safe.
// ---------------------------------------------------------------------------
extern "C" void kernel_launch(void* const* d_in, const int* in_sizes, int n_in,
                              void* d_out, int out_size, void* d_ws, size_t ws_size,
                              hipStream_t stream) {
    const float* x    = (const float*)d_in[0];   // [N, F]
    const int*   src  = (const int*)  d_in[1];   // [K, E]
    const int*   dst  = (const int*)  d_in[2];   // [K, E]
    const float* vals = (const float*)d_in[3];   // [K, E]
    const float* W    = (const float*)d_in[4];   // [OUT, K*F]
    const float* b    = (const float*)d_in[5];   // [OUT]
    float* out = (float*)d_out;                  // [N, OUT]
    float* y   = (float*)d_ws;                   // [N, F] scratch, reused per hop

    const int N = in_sizes[0] / FDIM;            // 100000
    const int E = in_sizes[1] / KHOP;            // 1600000

    // 1) out = broadcast(b)
    {
        int total4 = N * (ODIM / 4);
        int blocks = (total4 + 255) / 256;
        khop_init_out<<<blocks, 256, 0, stream>>>(out, b, total4);
    }

    // 2,3) per hop: dense fp32 WMMA GEMM, then edge scatter with fp32 atomics
    const int gemm_blocks    = N / 16;                    // 6250 (exact)
    const int scatter_blocks = (E * 32 + 255) / 256;      // one wave per edge
    for (int k = 0; k < KHOP; ++k) {
        khop_gemm_xWk<<<gemm_blocks, 256, 0, stream>>>(x, W, y, k);
        khop_scatter<<<scatter_blocks, 256, 0, stream>>>(
            src + (size_t)k * E, dst + (size_t)k * E, vals + (size_t)k * E,
            y, out, E);
    }
}